// LabelEncoder_40252433498602
// MI455X (gfx1250) — compile-verified
//
#include <hip/hip_runtime.h>
#include <cstdint>

#define NUM_CLASSES 80
#define KMATCH 10
#define EPSV 1e-9f

// ---------------------------------------------------------------------------
// CDNA5 async global->LDS helpers (gfx1250). Inline asm is used because the
// clang builtins for these differ across toolchains; asm is portable.
// global_load_async_to_lds_bN  VDST(lds byte addr vgpr), VADDR(offset vgpr),
//                              SADDR(sgpr64 base)      -- tracked by ASYNCcnt.
// ---------------------------------------------------------------------------
__device__ __forceinline__ unsigned lds_off(const void* p) {
    // Generic pointer to a __shared__ object: addr[31:0] is the LDS byte
    // address (LDS aperture maps by truncation per CDNA5 ISA 10.2).
    return (unsigned)(uintptr_t)p;
}
__device__ __forceinline__ void async_copy_b128(unsigned ldsaddr, unsigned voff,
                                                uint64_t sbase) {
    asm volatile("global_load_async_to_lds_b128 %0, %1, %2"
                 :: "v"(ldsaddr), "v"(voff), "s"(sbase) : "memory");
}
__device__ __forceinline__ void async_copy_b64(unsigned ldsaddr, unsigned voff,
                                               uint64_t sbase) {
    asm volatile("global_load_async_to_lds_b64 %0, %1, %2"
                 :: "v"(ldsaddr), "v"(voff), "s"(sbase) : "memory");
}
__device__ __forceinline__ void wait_async_le2() {
    asm volatile("s_wait_asynccnt 0x2" ::: "memory");
}
__device__ __forceinline__ void wait_async_0() {
    asm volatile("s_wait_asynccnt 0x0" ::: "memory");
}

// ---------------------------------------------------------------------------
// Pass 0: zero the scatter buffers (d_ws is poisoned by the harness).
// keys[b*A+a]  : u64 packed {iou_bits[31:0]<<32 | (0xFFFFFFFF - g)}
// normbits[..] : u32 float bits of norm (non-negative -> bit order == fp order)
// ---------------------------------------------------------------------------
__global__ void tal_init_kernel(unsigned long long* __restrict__ keys,
                                unsigned* __restrict__ normbits, int BA) {
    int t = blockIdx.x * blockDim.x + threadIdx.x;
    if (t < BA) { keys[t] = 0ull; normbits[t] = 0u; }
}

// ---------------------------------------------------------------------------
// Pass 1: one wave32 per (b,g). Stream anchors through async-LDS double
// buffer, keep per-lane sorted top-10, wave-merge to exact top-10, scatter.
// ---------------------------------------------------------------------------
__global__ __launch_bounds__(32) void tal_assign_kernel(
    const float* __restrict__ scores,          // (B,A,C)
    const float* __restrict__ dbox,            // (B,A,4)
    const float* __restrict__ anchors,         // (A,2)
    const int*   __restrict__ gt_labels,       // (B,G)
    const float* __restrict__ gt_bboxes,       // (B,G,4)
    const unsigned char* __restrict__ gt_mask, // (B,G) bool
    unsigned long long* __restrict__ keys,     // (B,A)
    unsigned* __restrict__ normbits,           // (B,A)
    int A, int G)
{
    const int bg = blockIdx.x;
    const int b = bg / G, g = bg % G;
    if (!gt_mask[b * G + g]) return;   // masked GT contributes nothing

    const int lane = threadIdx.x;
    const float gx1 = gt_bboxes[(b * G + g) * 4 + 0];
    const float gy1 = gt_bboxes[(b * G + g) * 4 + 1];
    const float gx2 = gt_bboxes[(b * G + g) * 4 + 2];
    const float gy2 = gt_bboxes[(b * G + g) * 4 + 3];
    const float area_g = (gx2 - gx1) * (gy2 - gy1);
    const int cls = gt_labels[b * G + g];
    const float* sb = scores + (size_t)b * A * NUM_CLASSES + cls;

    const uint64_t box_base = (uint64_t)(uintptr_t)(dbox + (size_t)b * A * 4);
    const uint64_t anc_base = (uint64_t)(uintptr_t)anchors;

    __shared__ float4 boxbuf[2][32];
    __shared__ float2 ancbuf[2][32];

    // per-lane sorted (descending) top-10; all indices static -> stays in VGPRs
    float tv[KMATCH]; int ti[KMATCH];
#pragma unroll
    for (int j = 0; j < KMATCH; ++j) { tv[j] = 0.0f; ti[j] = 0x7FFFFFFF; }

    const int nt = (A + 31) >> 5;
    { // prologue: tile 0 into buffer 0
        unsigned a0 = (unsigned)min(lane, A - 1);
        async_copy_b128(lds_off(&boxbuf[0][lane]), a0 * 16u, box_base);
        async_copy_b64 (lds_off(&ancbuf[0][lane]), a0 * 8u,  anc_base);
    }
    for (int t = 0; t < nt; ++t) {
        if (t + 1 < nt) { // issue next tile, then wait for current (<=2 left)
            unsigned an = (unsigned)min((t + 1) * 32 + lane, A - 1);
            int nb = (t + 1) & 1;
            async_copy_b128(lds_off(&boxbuf[nb][lane]), an * 16u, box_base);
            async_copy_b64 (lds_off(&ancbuf[nb][lane]), an * 8u,  anc_base);
            wait_async_le2();
        } else {
            wait_async_0();
        }
        const int a = t * 32 + lane;
        if (a < A) {
            float4 pb = boxbuf[t & 1][lane];
            float2 an = ancbuf[t & 1][lane];
            // strided score gather: prefetch next iteration's line
            __builtin_prefetch(sb + (size_t)min(a + 32, A - 1) * NUM_CLASSES, 0, 1);
            float ltx = fmaxf(gx1, pb.x), lty = fmaxf(gy1, pb.y);
            float rbx = fminf(gx2, pb.z), rby = fminf(gy2, pb.w);
            float w = fmaxf(rbx - ltx, 0.f), h = fmaxf(rby - lty, 0.f);
            float inter  = w * h;
            float area_p = (pb.z - pb.x) * (pb.w - pb.y);
            float iou = inter / (area_g + area_p - inter + EPSV);
            bool inb = (an.x > gx1) && (an.x < gx2) && (an.y > gy1) && (an.y < gy2);
            if (inb && iou > 0.f) {
                float sc = sb[(size_t)a * NUM_CLASSES];
                float i2 = iou * iou, i6 = i2 * i2 * i2;
                float align = sqrtf(sc) * i6;      // score^0.5 * iou^6
                if (align > tv[KMATCH - 1]) {      // insert (unrolled bubble)
                    tv[KMATCH - 1] = align; ti[KMATCH - 1] = a;
#pragma unroll
                    for (int j = KMATCH - 1; j > 0; --j) {
                        if (tv[j] > tv[j - 1]) {
                            float fv = tv[j]; tv[j] = tv[j - 1]; tv[j - 1] = fv;
                            int   iv = ti[j]; ti[j] = ti[j - 1]; ti[j - 1] = iv;
                        }
                    }
                }
            }
        }
    }

    // wave32 merge: 10 rounds of argmax (ties -> smallest anchor index,
    // matching jax.lax.top_k's stable ordering). Round index r is a
    // compile-time constant, so lane r captures round r's winner as scalars:
    // no private arrays, no dynamic indexing, no scratch.
    float max_align = 0.f;     // round-0 winner (wave-uniform)
    float myAlign = 0.f;       // lane r holds the r-th top value
    int   myA = 0;             // ... and its anchor index
#pragma unroll
    for (int r = 0; r < KMATCH; ++r) {
        float v = tv[0]; int i = ti[0];
#pragma unroll
        for (int off = 16; off > 0; off >>= 1) {
            float ov = __shfl_xor(v, off, 32);
            int   oi = __shfl_xor(i, off, 32);
            if (ov > v || (ov == v && oi < i)) { v = ov; i = oi; }
        }
        if (r == 0) max_align = v;                // identical on all lanes
        if (lane == r) { myAlign = v; myA = i; }  // static capture
        if (i == ti[0]) {                         // winner pops its head
#pragma unroll
            for (int j = 0; j < KMATCH - 1; ++j) { tv[j] = tv[j + 1]; ti[j] = ti[j + 1]; }
            tv[KMATCH - 1] = 0.0f; ti[KMATCH - 1] = 0x7FFFFFFF;
        }
    }

    if (max_align <= 0.f) return;                 // no valid matches for this gt

    // lanes 0..9 recompute overlap at their matched anchor
    const bool valid = (lane < KMATCH) && (myAlign > 0.f);
    float ov = 0.f;
    if (valid) {
        float4 pb = ((const float4*)dbox)[(size_t)b * A + myA];
        float ltx = fmaxf(gx1, pb.x), lty = fmaxf(gy1, pb.y);
        float rbx = fminf(gx2, pb.z), rby = fminf(gy2, pb.w);
        float w = fmaxf(rbx - ltx, 0.f), h = fmaxf(rby - lty, 0.f);
        float inter  = w * h;
        float area_p = (pb.z - pb.x) * (pb.w - pb.y);
        ov = inter / (area_g + area_p - inter + EPSV);
    }
    float mo = ov;                                // max_ov over matched anchors
#pragma unroll
    for (int off = 16; off > 0; off >>= 1) mo = fmaxf(mo, __shfl_xor(mo, off, 32));

    if (valid) {
        // argmax over g with first-index tie-break: pack {iou_bits, ~g}
        unsigned long long key =
            ((unsigned long long)__float_as_uint(ov) << 32) |
            (unsigned long long)(0xFFFFFFFFu - (unsigned)g);
        atomicMax(&keys[(size_t)b * A + myA], key);
        float nv = myAlign * mo / (max_align + EPSV);
        atomicMax(&normbits[(size_t)b * A + myA], __float_as_uint(nv));
    }
}

// ---------------------------------------------------------------------------
// Pass 2: bbox_labels (B,A,4) + fg (B,A) from the packed keys.
// ---------------------------------------------------------------------------
__global__ void tal_bbox_fg_kernel(const unsigned long long* __restrict__ keys,
                                   const float* __restrict__ gt_bboxes,
                                   float* __restrict__ out_bbox,
                                   float* __restrict__ out_fg,
                                   int BA, int A, int G) {
    int t = blockIdx.x * blockDim.x + threadIdx.x;
    if (t >= BA) return;
    unsigned long long key = keys[t];
    bool fg = (unsigned)(key >> 32) != 0u;        // matched iff best iou > 0
    float4 bb = make_float4(-1.f, -1.f, -1.f, -1.f);
    if (fg) {
        int b = t / A;
        int g = (int)(0xFFFFFFFFu - (unsigned)(key & 0xFFFFFFFFull));
        bb = ((const float4*)gt_bboxes)[(size_t)b * G + g];
    }
    ((float4*)out_bbox)[t] = bb;
    out_fg[t] = fg ? 1.0f : 0.0f;
}

// ---------------------------------------------------------------------------
// Pass 3: class_labels (B,A,C) = one_hot(cls) * norm. Bandwidth-dominant
// write (86 MB) -> coalesced float4 stores, 4 classes per thread.
// ---------------------------------------------------------------------------
__global__ void tal_class_kernel(const unsigned long long* __restrict__ keys,
                                 const unsigned* __restrict__ normbits,
                                 const int* __restrict__ gt_labels,
                                 float* __restrict__ out_cls,
                                 int BA, int A, int G) {
    const int Q = NUM_CLASSES / 4;                // 20 float4 per (b,a)
    int t = blockIdx.x * blockDim.x + threadIdx.x;
    if (t >= BA * Q) return;
    int ba = t / Q;
    int q  = t - ba * Q;
    unsigned long long key = keys[ba];
    float4 v = make_float4(0.f, 0.f, 0.f, 0.f);
    if ((unsigned)(key >> 32) != 0u) {
        int b = ba / A;
        int g = (int)(0xFFFFFFFFu - (unsigned)(key & 0xFFFFFFFFull));
        int cls = gt_labels[b * G + g];
        if ((cls >> 2) == q) {
            float nv = __uint_as_float(normbits[ba]);
            ((float*)&v)[cls & 3] = nv;
        }
    }
    ((float4*)out_cls)[t] = v;
}

// ---------------------------------------------------------------------------
extern "C" void kernel_launch(void* const* d_in, const int* in_sizes, int n_in,
                              void* d_out, int out_size, void* d_ws, size_t ws_size,
                              hipStream_t stream) {
    const float*         scores    = (const float*)d_in[0];
    const float*         dbox      = (const float*)d_in[1];
    const float*         anchors   = (const float*)d_in[2];
    const int*           gt_labels = (const int*)d_in[3];
    const float*         gt_bboxes = (const float*)d_in[4];
    const unsigned char* gt_mask   = (const unsigned char*)d_in[5];

    const int A  = in_sizes[2] / 2;                       // 8400
    const int B  = in_sizes[0] / (A * NUM_CLASSES);       // 32
    const int G  = in_sizes[3] / B;                       // 100
    const int BA = B * A;

    unsigned long long* keys = (unsigned long long*)d_ws;            // BA*8 B
    unsigned* normbits = (unsigned*)((char*)d_ws + (size_t)BA * 8);  // BA*4 B

    float* out      = (float*)d_out;
    float* out_bbox = out;                                  // (B,A,4)
    float* out_cls  = out + (size_t)BA * 4;                 // (B,A,C)
    float* out_fg   = out + (size_t)BA * 4 + (size_t)BA * NUM_CLASSES; // (B,A)

    tal_init_kernel<<<(BA + 255) / 256, 256, 0, stream>>>(keys, normbits, BA);
    tal_assign_kernel<<<B * G, 32, 0, stream>>>(scores, dbox, anchors, gt_labels,
                                                gt_bboxes, gt_mask, keys, normbits,
                                                A, G);
    tal_bbox_fg_kernel<<<(BA + 255) / 256, 256, 0, stream>>>(keys, gt_bboxes,
                                                             out_bbox, out_fg,
                                                             BA, A, G);
    int totalC = BA * (NUM_CLASSES / 4);
    tal_class_kernel<<<(totalC + 255) / 256, 256, 0, stream>>>(keys, normbits,
                                                               gt_labels, out_cls,
                                                               BA, A, G);
}